// DiffDMC_41601053229822
// MI455X (gfx1250) — compile-verified
//
#include <hip/hip_runtime.h>
#include <hip/hip_bf16.h>

// ---------------------------------------------------------------------------
// DiffDMC forward for D=128, ISO=0.0 on gfx1250 (MI455X).
// Store-bound (~195 MB traffic, ~0.2 GFLOP) -> optimize data movement:
//  * verts kernel: 4x4x16-cell tiles, corner grid+deform staged into LDS via
//    CDNA5 async global->LDS copies (ASYNCcnt path), compute reads from LDS.
//  * quads kernel: pure index math, one 16B-aligned b128 NT store per quad.
//  * all outputs (161 MB) stored with TH=NT so the 33.5 MB of inputs stay
//    resident in the 192 MB L2.
// ---------------------------------------------------------------------------

#define DD   128                 // input grid resolution
#define NC   129                 // cells per axis of padded grid (130-1)

static constexpr int NC3       = NC * NC * NC;            // 2,146,689 dual cells
static constexpr int NQ_AXIS   = NC * (NC - 1) * (NC - 1);// 2,113,536 (per axis, =256*8256)
static constexpr int NQ        = 3 * NQ_AXIS;             // 6,340,608 quads

// Flat float offsets into d_out (verts, vert_mask, quads, qmask concatenated)
static constexpr int OFF_VERTS = 0;
static constexpr int OFF_VMASK = 3 * NC3;                 //  6,440,067
static constexpr int OFF_QUADS = OFF_VMASK + NC3;         //  8,586,756 (16B aligned)
static constexpr int OFF_QMASK = OFF_QUADS + 4 * NQ;      // 33,949,188

typedef __attribute__((ext_vector_type(4))) float v4f;

#if __has_builtin(__builtin_amdgcn_global_load_async_to_lds_b32)
#define HAS_ASYNC_LDS 1
#else
#define HAS_ASYNC_LDS 0
#endif

typedef __attribute__((address_space(1))) int* gint_p;   // global dword ptr
typedef __attribute__((address_space(3))) int* lint_p;   // LDS dword ptr

// Async copy of one dword global->LDS (ASYNCcnt-tracked on CDNA5); fallback is
// a synchronous load + ds_store with identical semantics.
__device__ __forceinline__ void g2l_b32(const float* g, float* l) {
#if HAS_ASYNC_LDS
    __builtin_amdgcn_global_load_async_to_lds_b32((gint_p)(g), (lint_p)(l), 0, 0);
#else
    *l = *g;
#endif
}

__device__ __forceinline__ void wait_async_then_barrier() {
#if HAS_ASYNC_LDS
#if __has_builtin(__builtin_amdgcn_s_wait_asynccnt)
    __builtin_amdgcn_s_wait_asynccnt(0);
#else
    asm volatile("s_wait_asynccnt 0" ::: "memory");
#endif
#endif
    __syncthreads();
}

// Padded grid sample (used by the quads kernel): pad value = iso+1 = 1.0
__device__ __forceinline__ float gval(const float* __restrict__ grid, int x, int y, int z) {
    unsigned ux = (unsigned)(x - 1), uy = (unsigned)(y - 1), uz = (unsigned)(z - 1);
    if (ux < (unsigned)DD && uy < (unsigned)DD && uz < (unsigned)DD)
        return grid[((size_t)ux * DD + uy) * DD + uz];
    return 1.0f;
}

__device__ __forceinline__ int lin129(int i, int j, int k) { return (i * NC + j) * NC + k; }

// ---------------------------------------------------------------------------
// Kernel 1: dual vertices + vertex mask.
// Block = 256 threads = 4(i) x 4(j) x 16(k) cells. Corner footprint 5x5x17 is
// staged into LDS (grid: 425 f32, deform: 1275 f32 -> 6.8 KB/block).
// Grid launch: (9 k-tiles, 33 j-tiles, 33 i-tiles).
// ---------------------------------------------------------------------------
#define TSX 425                 // 5*5*17 staged corner sites
__global__ __launch_bounds__(256) void dmc_verts_kernel(const float* __restrict__ grid,
                                                        const float* __restrict__ deform,
                                                        float* __restrict__ out) {
    __shared__ float gs[TSX];        // padded grid corner values
    __shared__ float dls[TSX * 3];   // padded deform corner values

    const int tid = threadIdx.x;
    const int x0 = 4 * blockIdx.z, y0 = 4 * blockIdx.y, z0 = 16 * blockIdx.x;

    // ---- stage grid corners (pad = 1.0 outside the 128^3 volume) ----
    for (int e = tid; e < TSX; e += 256) {
        int ex = e / 85, r = e - ex * 85, ey = r / 17, ez = r - ey * 17;
        unsigned ux = (unsigned)(x0 + ex - 1), uy = (unsigned)(y0 + ey - 1),
                 uz = (unsigned)(z0 + ez - 1);
        if (ux < (unsigned)DD && uy < (unsigned)DD && uz < (unsigned)DD)
            g2l_b32(grid + ((size_t)ux * DD + uy) * DD + uz, gs + e);
        else
            gs[e] = 1.0f;
    }
    // ---- stage deform corners (pad = 0) ----
    for (int e = tid; e < TSX; e += 256) {
        int ex = e / 85, r = e - ex * 85, ey = r / 17, ez = r - ey * 17;
        unsigned ux = (unsigned)(x0 + ex - 1), uy = (unsigned)(y0 + ey - 1),
                 uz = (unsigned)(z0 + ez - 1);
        if (ux < (unsigned)DD && uy < (unsigned)DD && uz < (unsigned)DD) {
            const float* dp = deform + (((size_t)ux * DD + uy) * DD + uz) * 3u;
            g2l_b32(dp + 0, dls + 3 * e + 0);
            g2l_b32(dp + 1, dls + 3 * e + 1);
            g2l_b32(dp + 2, dls + 3 * e + 2);
        } else {
            dls[3 * e + 0] = 0.0f;
            dls[3 * e + 1] = 0.0f;
            dls[3 * e + 2] = 0.0f;
        }
    }
    wait_async_then_barrier();

    const int lk = tid & 15, lj = (tid >> 4) & 3, li = tid >> 6;
    const int i = x0 + li, j = y0 + lj, k = z0 + lk;
    if (i >= NC || j >= NC || k >= NC) return;   // partial tiles (after barrier)

    const int cdx[8] = {0, 1, 1, 0, 0, 1, 1, 0};
    const int cdy[8] = {0, 0, 1, 1, 0, 0, 1, 1};
    const int cdz[8] = {0, 0, 0, 0, 1, 1, 1, 1};

    float v[8], px[8], py[8], pz[8];
#pragma unroll
    for (int c = 0; c < 8; ++c) {
        int s = ((li + cdx[c]) * 5 + (lj + cdy[c])) * 17 + (lk + cdz[c]);
        v[c]  = gs[s];
        px[c] = (float)(i + cdx[c]) + dls[3 * s + 0];
        py[c] = (float)(j + cdy[c]) + dls[3 * s + 1];
        pz[c] = (float)(k + cdz[c]) + dls[3 * s + 2];
    }

    const int ea[12] = {0, 1, 2, 3, 4, 5, 6, 7, 0, 1, 2, 3};
    const int eb[12] = {1, 2, 3, 0, 5, 6, 7, 4, 4, 5, 6, 7};

    float sx = 0.0f, sy = 0.0f, sz = 0.0f, cnt = 0.0f;
#pragma unroll
    for (int e = 0; e < 12; ++e) {
        int a = ea[e], b = eb[e];
        float va = v[a], vb = v[b];
        if ((va < 0.0f) != (vb < 0.0f)) {           // sign-change classification (iso=0)
            float denom = vb - va;
            float t = (0.0f - va) / (denom == 0.0f ? 1.0f : denom);
            t = fminf(fmaxf(t, 0.0f), 1.0f);
            sx += px[a] + t * (px[b] - px[a]);
            sy += py[a] + t * (py[b] - py[a]);
            sz += pz[a] + t * (pz[b] - pz[a]);
            cnt += 1.0f;
        }
    }

    bool active = cnt > 0.0f;
    float inv = 1.0f / fmaxf(cnt, 1.0f);
    const float dims = (float)(DD - 1);             // 127.0
    float ox = active ? (sx * inv - 1.0f) / dims : 0.0f;
    float oy = active ? (sy * inv - 1.0f) / dims : 0.0f;
    float oz = active ? (sz * inv - 1.0f) / dims : 0.0f;

    const int n = (i * NC + j) * NC + k;
    float* vo = out + OFF_VERTS + 3 * (size_t)n;    // streamed once -> TH=NT
    __builtin_nontemporal_store(ox, vo + 0);
    __builtin_nontemporal_store(oy, vo + 1);
    __builtin_nontemporal_store(oz, vo + 2);
    __builtin_nontemporal_store(active ? 1.0f : 0.0f, out + OFF_VMASK + n);
}

// ---------------------------------------------------------------------------
// Kernel 2: quads (constant connectivity, exact in f32 since idx < 2^24) +
// qmask (sign-change across interior grid edges). One thread per quad.
// Each axis region = 2,113,536 = 256*8256 -> 256-thread blocks never diverge
// on the axis branch.
// ---------------------------------------------------------------------------
__global__ __launch_bounds__(256) void dmc_quads_kernel(const float* __restrict__ grid,
                                                        float* __restrict__ out) {
    int q = blockIdx.x * 256 + threadIdx.x;
    if (q >= NQ) return;

    const int E = NC - 1;                           // 128
    int q0, q1, q2, q3;
    bool mask;

    if (q < NQ_AXIS) {                              // axis-0 edges: ii in [0,129), jj,kk in [1,129)
        int t = q;
        int ii = t / (E * E);
        int r  = t - ii * (E * E);
        int jj = 1 + (r / E);
        int kk = 1 + (r - (r / E) * E);
        q0 = lin129(ii, jj - 1, kk - 1);
        q1 = lin129(ii, jj,     kk - 1);
        q2 = lin129(ii, jj,     kk);
        q3 = lin129(ii, jj - 1, kk);
        mask = (gval(grid, ii, jj, kk) < 0.0f) != (gval(grid, ii + 1, jj, kk) < 0.0f);
    } else if (q < 2 * NQ_AXIS) {                   // axis-1 edges: ii,kk in [1,129), jj in [0,129)
        int t = q - NQ_AXIS;
        int ii = 1 + t / (NC * E);
        int r  = t - (ii - 1) * (NC * E);
        int jj = r / E;
        int kk = 1 + (r - jj * E);
        q0 = lin129(ii - 1, jj, kk - 1);
        q1 = lin129(ii,     jj, kk - 1);
        q2 = lin129(ii,     jj, kk);
        q3 = lin129(ii - 1, jj, kk);
        mask = (gval(grid, ii, jj, kk) < 0.0f) != (gval(grid, ii, jj + 1, kk) < 0.0f);
    } else {                                        // axis-2 edges: ii,jj in [1,129), kk in [0,129)
        int t = q - 2 * NQ_AXIS;
        int ii = 1 + t / (E * NC);
        int r  = t - (ii - 1) * (E * NC);
        int jj = 1 + (r / NC);
        int kk = r - (r / NC) * NC;
        q0 = lin129(ii - 1, jj - 1, kk);
        q1 = lin129(ii,     jj - 1, kk);
        q2 = lin129(ii,     jj,     kk);
        q3 = lin129(ii - 1, jj,     kk);
        mask = (gval(grid, ii, jj, kk) < 0.0f) != (gval(grid, ii, jj, kk + 1) < 0.0f);
    }

    // One 16B-aligned b128 NT store per quad (101 MB stream; keep L2 for inputs).
    v4f qv = {(float)q0, (float)q1, (float)q2, (float)q3};
    __builtin_nontemporal_store(qv, (v4f*)(out + OFF_QUADS) + q);
    __builtin_nontemporal_store(mask ? 1.0f : 0.0f, out + OFF_QMASK + q);
}

// ---------------------------------------------------------------------------
extern "C" void kernel_launch(void* const* d_in, const int* in_sizes, int n_in,
                              void* d_out, int out_size, void* d_ws, size_t ws_size,
                              hipStream_t stream) {
    (void)in_sizes; (void)n_in; (void)out_size; (void)d_ws; (void)ws_size;
    const float* grid   = (const float*)d_in[0];   // (128,128,128) f32
    const float* deform = (const float*)d_in[1];   // (128,128,128,3) f32
    float* out = (float*)d_out;

    dim3 vgrid(9, 33, 33);                          // k-tiles(16), j-tiles(4), i-tiles(4)
    dmc_verts_kernel<<<vgrid, 256, 0, stream>>>(grid, deform, out);
    dmc_quads_kernel<<<NQ / 256, 256, 0, stream>>>(grid, out);
}